// Model_42253888258528
// MI455X (gfx1250) — compile-verified
//
#include <hip/hip_runtime.h>

// 5x5 median filter, 2048x2048 fp32, reflect padding.
// CDNA5 path: async global->LDS staging (ASYNCcnt) + wave32 min/max network.

#define IMG_H 2048
#define IMG_W 2048
#define TILE_W 64
#define TILE_H 16
#define RAD 2
#define LPITCH (TILE_W + 2 * RAD)          // 68
#define LROWS  (TILE_H + 2 * RAD)          // 20
#define LDS_ELEMS (LPITCH * LROWS)         // 1360
#define NTHREADS 256

__device__ __forceinline__ int reflect_idx(int i, int n) {
    i = (i < 0) ? -i : i;                  // left mirror (no edge repeat)
    i = (i >= n) ? (2 * n - 2 - i) : i;    // right mirror
    return i;
}

// compare-exchange: a = min, b = max (inputs are finite normals; no NaNs)
#define CSWAP(a, b) { float _t = fminf((a), (b)); (b) = fmaxf((a), (b)); (a) = _t; }

__global__ __launch_bounds__(NTHREADS) void median5x5_kernel(
    const float* __restrict__ img, float* __restrict__ out) {
    __shared__ float tile[LDS_ELEMS];

    const int tid = threadIdx.x;
    const int bx = blockIdx.x;             // 0..31
    const int by = blockIdx.y;             // 0..127
    const int gx0 = bx * TILE_W - RAD;
    const int gy0 = by * TILE_H - RAD;

    // ---- Stage reflect-padded (20 x 68) tile into LDS via async DMA loads ----
    for (int idx = tid; idx < LDS_ELEMS; idx += NTHREADS) {
        const int ly = idx / LPITCH;
        const int lx = idx - ly * LPITCH;
        const int gy = reflect_idx(gy0 + ly, IMG_H);
        const int gx = reflect_idx(gx0 + lx, IMG_W);
        unsigned long long gaddr =
            (unsigned long long)(const void*)(img + (size_t)gy * IMG_W + gx);
        unsigned lds_addr = (unsigned)(unsigned long long)(const void*)&tile[idx];
        // global memory -> LDS, no VGPR round-trip; tracked by ASYNCcnt
        asm volatile("global_load_async_to_lds_b32 %0, %1, off"
                     :: "v"(lds_addr), "v"(gaddr)
                     : "memory");
    }
    asm volatile("s_wait_asynccnt 0" ::: "memory"); // this wave's async writes done
    __syncthreads();                                // all waves' writes visible

    // ---- Each thread computes 4 output pixels ----
    const int tx = tid & (TILE_W - 1);     // 0..63
    const int ty = tid >> 6;               // 0..3

    #pragma unroll
    for (int k = 0; k < 4; ++k) {
        const int row = ty + 4 * k;        // 0..15

        float p[25];
        #pragma unroll
        for (int dy = 0; dy < 5; ++dy) {
            #pragma unroll
            for (int dx = 0; dx < 5; ++dx) {
                p[dy * 5 + dx] = tile[(row + dy) * LPITCH + (tx + dx)];
            }
        }

        // Devillard opt_med25: 99-exchange exact median-of-25 network
        CSWAP(p[0],p[1]);   CSWAP(p[3],p[4]);   CSWAP(p[2],p[4]);
        CSWAP(p[2],p[3]);   CSWAP(p[6],p[7]);   CSWAP(p[5],p[7]);
        CSWAP(p[5],p[6]);   CSWAP(p[9],p[10]);  CSWAP(p[8],p[10]);
        CSWAP(p[8],p[9]);   CSWAP(p[12],p[13]); CSWAP(p[11],p[13]);
        CSWAP(p[11],p[12]); CSWAP(p[15],p[16]); CSWAP(p[14],p[16]);
        CSWAP(p[14],p[15]); CSWAP(p[18],p[19]); CSWAP(p[17],p[19]);
        CSWAP(p[17],p[18]); CSWAP(p[21],p[22]); CSWAP(p[20],p[22]);
        CSWAP(p[20],p[21]); CSWAP(p[23],p[24]); CSWAP(p[2],p[5]);
        CSWAP(p[3],p[6]);   CSWAP(p[0],p[6]);   CSWAP(p[0],p[3]);
        CSWAP(p[4],p[7]);   CSWAP(p[1],p[7]);   CSWAP(p[1],p[4]);
        CSWAP(p[11],p[14]); CSWAP(p[8],p[14]);  CSWAP(p[8],p[11]);
        CSWAP(p[12],p[15]); CSWAP(p[9],p[15]);  CSWAP(p[9],p[12]);
        CSWAP(p[13],p[16]); CSWAP(p[10],p[16]); CSWAP(p[10],p[13]);
        CSWAP(p[20],p[23]); CSWAP(p[17],p[23]); CSWAP(p[17],p[20]);
        CSWAP(p[21],p[24]); CSWAP(p[18],p[24]); CSWAP(p[18],p[21]);
        CSWAP(p[19],p[22]); CSWAP(p[8],p[17]);  CSWAP(p[9],p[18]);
        CSWAP(p[0],p[18]);  CSWAP(p[0],p[9]);   CSWAP(p[10],p[19]);
        CSWAP(p[1],p[19]);  CSWAP(p[1],p[10]);  CSWAP(p[11],p[20]);
        CSWAP(p[2],p[20]);  CSWAP(p[2],p[11]);  CSWAP(p[12],p[21]);
        CSWAP(p[3],p[21]);  CSWAP(p[3],p[12]);  CSWAP(p[13],p[22]);
        CSWAP(p[4],p[22]);  CSWAP(p[4],p[13]);  CSWAP(p[14],p[23]);
        CSWAP(p[5],p[23]);  CSWAP(p[5],p[14]);  CSWAP(p[15],p[24]);
        CSWAP(p[6],p[24]);  CSWAP(p[6],p[15]);  CSWAP(p[7],p[16]);
        CSWAP(p[7],p[19]);  CSWAP(p[13],p[21]); CSWAP(p[15],p[23]);
        CSWAP(p[7],p[13]);  CSWAP(p[7],p[15]);  CSWAP(p[1],p[9]);
        CSWAP(p[3],p[11]);  CSWAP(p[5],p[17]);  CSWAP(p[11],p[17]);
        CSWAP(p[9],p[17]);  CSWAP(p[4],p[10]);  CSWAP(p[6],p[12]);
        CSWAP(p[7],p[14]);  CSWAP(p[4],p[6]);   CSWAP(p[4],p[7]);
        CSWAP(p[12],p[14]); CSWAP(p[10],p[14]); CSWAP(p[6],p[7]);
        CSWAP(p[10],p[12]); CSWAP(p[6],p[10]);  CSWAP(p[6],p[17]);
        CSWAP(p[12],p[17]); CSWAP(p[7],p[17]);  CSWAP(p[7],p[10]);
        CSWAP(p[12],p[18]); CSWAP(p[7],p[12]);  CSWAP(p[10],p[18]);
        CSWAP(p[12],p[20]); CSWAP(p[10],p[20]); CSWAP(p[10],p[12]);

        const int gy = by * TILE_H + row;
        const int gx = bx * TILE_W + tx;
        out[(size_t)gy * IMG_W + gx] = p[12];
    }
}

extern "C" void kernel_launch(void* const* d_in, const int* in_sizes, int n_in,
                              void* d_out, int out_size, void* d_ws, size_t ws_size,
                              hipStream_t stream) {
    (void)in_sizes; (void)n_in; (void)out_size; (void)d_ws; (void)ws_size;
    const float* img = (const float*)d_in[0];
    float* out = (float*)d_out;
    dim3 grid(IMG_W / TILE_W, IMG_H / TILE_H);  // 32 x 128
    dim3 block(NTHREADS);
    median5x5_kernel<<<grid, block, 0, stream>>>(img, out);
}